// nalu_transform_5488968204577
// MI455X (gfx1250) — compile-verified
//
#include <hip/hip_runtime.h>
#include <math.h>

#define BB 128
#define DD 1024
#define OO 512
#define THREE_O 1536
#define EPS_LN 1e-5f
#define NCHUNK 4   // d-dimension split for the attention kernel

typedef __attribute__((ext_vector_type(2))) float v2f;
typedef __attribute__((ext_vector_type(8))) float v8f;

// ---------------------------------------------------------------------------
// Elementwise prep: wa = tanh(wt) * sigmoid(mt)   over [D,O]
// ---------------------------------------------------------------------------
__global__ void prep_wa_kernel(const float* __restrict__ wt,
                               const float* __restrict__ mt,
                               float* __restrict__ wa, int n) {
  int i = blockIdx.x * blockDim.x + threadIdx.x;
  if (i < n) {
    float s = 1.f / (1.f + __expf(-mt[i]));
    wa[i] = tanhf(wt[i]) * s;
  }
}

// ax = asinh(x) over [B,D]
__global__ void prep_ax_kernel(const float* __restrict__ x,
                               float* __restrict__ ax, int n) {
  int i = blockIdx.x * blockDim.x + threadIdx.x;
  if (i < n) ax[i] = asinhf(x[i]);
}

// ---------------------------------------------------------------------------
// Scalar reductions for the rank-1 attention collapse:
//   scal[0] = alpha = sum_o kq[o]*kk[o]
//   scal[1] = beta  = sum_o bq[o]*kk[o]
//   scal[2] = E     = sum_d ek[d]
// ---------------------------------------------------------------------------
__global__ __launch_bounds__(256) void scalars_kernel(
    const float* __restrict__ embk, const float* __restrict__ embb,
    const float* __restrict__ ek, float* __restrict__ scal) {
  __shared__ float ra[256], rb[256], re[256];
  const int tid = threadIdx.x;
  const float* kq = embk;
  const float* kk = embk + OO;
  const float* bq = embb;
  float a = 0.f, b = 0.f, e = 0.f;
  for (int o = tid; o < OO; o += 256) {
    a = fmaf(kq[o], kk[o], a);
    b = fmaf(bq[o], kk[o], b);
  }
  for (int d = tid; d < DD; d += 256) e += ek[d];
  ra[tid] = a; rb[tid] = b; re[tid] = e;
  __syncthreads();
  for (int s = 128; s > 0; s >>= 1) {
    if (tid < s) { ra[tid] += ra[tid+s]; rb[tid] += rb[tid+s]; re[tid] += re[tid+s]; }
    __syncthreads();
  }
  if (tid == 0) { scal[0] = ra[0]; scal[1] = rb[0]; scal[2] = re[0]; }
}

// ---------------------------------------------------------------------------
// Fused 4-way WMMA GEMM over shared K=D:
//   acc1 = x@k1, accg = x@gt, acca = x@wa, accm = asinh(x)@wa
// Epilogue: feat[:,0:O]  = acc1 + b1
//           feat[:,O:2O] = g*a + (1-g)*m  with g=sigmoid(accg), m=sinh(accm)
// One wave32 per 16x16 output tile; V_WMMA_F32_16X16X4_F32, K stepped by 4.
// ---------------------------------------------------------------------------
__global__ __launch_bounds__(32) void gemm_fused_kernel(
    const float* __restrict__ x, const float* __restrict__ ax,
    const float* __restrict__ k1, const float* __restrict__ gt,
    const float* __restrict__ wa, const float* __restrict__ b1,
    float* __restrict__ feat) {
  const int lane = threadIdx.x;
  const int r = lane & 15;
  const int h = lane >> 4;
  const int row = blockIdx.y * 16 + r;   // A row (M)
  const int col = blockIdx.x * 16 + r;   // B col (N)
  v8f acc1 = {}; v8f accg = {}; v8f acca = {}; v8f accm = {};
  for (int k0 = 0; k0 < DD; k0 += 4) {
    const int ka = k0 + 2 * h;
    v2f aX = *(const v2f*)(x  + (size_t)row * DD + ka);
    v2f aA = *(const v2f*)(ax + (size_t)row * DD + ka);
    v2f b1v, bgv, bwv;
    b1v.x = k1[(size_t)(ka + 0) * OO + col];
    b1v.y = k1[(size_t)(ka + 1) * OO + col];
    bgv.x = gt[(size_t)(ka + 0) * OO + col];
    bgv.y = gt[(size_t)(ka + 1) * OO + col];
    bwv.x = wa[(size_t)(ka + 0) * OO + col];
    bwv.y = wa[(size_t)(ka + 1) * OO + col];
    acc1 = __builtin_amdgcn_wmma_f32_16x16x4_f32(false, aX, false, b1v, (short)0, acc1, false, false);
    accg = __builtin_amdgcn_wmma_f32_16x16x4_f32(false, aX, false, bgv, (short)0, accg, false, false);
    acca = __builtin_amdgcn_wmma_f32_16x16x4_f32(false, aX, false, bwv, (short)0, acca, false, false);
    accm = __builtin_amdgcn_wmma_f32_16x16x4_f32(false, aA, false, bwv, (short)0, accm, false, false);
  }
  for (int i = 0; i < 8; ++i) {
    const int m = blockIdx.y * 16 + i + 8 * h;
    const int n = blockIdx.x * 16 + r;
    const float c1 = acc1[i] + b1[n];
    const float g  = 1.f / (1.f + __expf(-accg[i]));
    const float a  = acca[i];
    const float mm = sinhf(accm[i]);
    feat[(size_t)m * THREE_O + n]      = c1;
    feat[(size_t)m * THREE_O + OO + n] = fmaf(g, a - mm, mm);  // g*a + (1-g)*m
  }
}

// ---------------------------------------------------------------------------
// Rank-1-collapsed attention. Grid = (B, NCHUNK); each block owns 256 d-rows
// (one per thread) of batch row b and computes a partial
//   Tpart[b*NCHUNK+chunk] = sum_{d in chunk} ek[d] * softmax_e(c_d*x[b,e]).x[b,:]
// with c_d = scale*(alpha*x[b,d]+beta).  x[b,:] cached in LDS; stable shift.
// 4096 waves total; e-loop unrolled 4x -> 4 independent exp/FMA chains/lane.
// ---------------------------------------------------------------------------
__global__ __launch_bounds__(256) void attn_T_kernel(
    const float* __restrict__ x, const float* __restrict__ ek,
    const float* __restrict__ scal, float* __restrict__ Tpart) {
  __shared__ float xs[DD];
  __shared__ float rmax[256];
  __shared__ float rmin[256];
  __shared__ float racc[256];
  __shared__ float s_xmax, s_xmin;
  const int b = blockIdx.x;
  const int chunk = blockIdx.y;
  const int tid = threadIdx.x;
  float mx = -3.402823466e38f, mn = 3.402823466e38f;
  for (int i = tid; i < DD; i += 256) {
    float v = x[(size_t)b * DD + i];
    xs[i] = v;
    mx = fmaxf(mx, v);
    mn = fminf(mn, v);
  }
  rmax[tid] = mx; rmin[tid] = mn;
  __syncthreads();
  for (int s = 128; s > 0; s >>= 1) {
    if (tid < s) {
      rmax[tid] = fmaxf(rmax[tid], rmax[tid + s]);
      rmin[tid] = fminf(rmin[tid], rmin[tid + s]);
    }
    __syncthreads();
  }
  if (tid == 0) { s_xmax = rmax[0]; s_xmin = rmin[0]; }
  __syncthreads();
  const float xmax = s_xmax, xmin = s_xmin;
  const float alpha = scal[0], beta = scal[1];
  const float scale = rsqrtf((float)OO);

  const int d = chunk * 256 + tid;          // one d-row per thread
  const float c = scale * fmaf(alpha, xs[d], beta);
  const float shift = (c >= 0.f) ? c * xmax : c * xmin;
  float den = 0.f, num = 0.f;
#pragma unroll 4
  for (int e = 0; e < DD; ++e) {
    const float xe = xs[e];
    const float z = __expf(fmaf(c, xe, -shift));
    den += z;
    num = fmaf(z, xe, num);
  }
  racc[tid] = ek[d] * (num / den);
  __syncthreads();
  for (int s = 128; s > 0; s >>= 1) {
    if (tid < s) racc[tid] += racc[tid + s];
    __syncthreads();
  }
  if (tid == 0) Tpart[b * NCHUNK + chunk] = racc[0];
}

// ---------------------------------------------------------------------------
// c4: s[b,o] = kv[o]*T[b] + bv[o]*E; layernorm over o; write feat[:,2O:3O]
// T[b] is the sum of the NCHUNK partials from attn_T_kernel.
// ---------------------------------------------------------------------------
__global__ __launch_bounds__(256) void c4_kernel(
    const float* __restrict__ embk, const float* __restrict__ embb,
    const float* __restrict__ scal, const float* __restrict__ Tpart,
    const float* __restrict__ ng, const float* __restrict__ nb,
    float* __restrict__ feat) {
  __shared__ float sv[OO];
  __shared__ float red[256];
  __shared__ float s_u, s_inv;
  const int b = blockIdx.x;
  const int tid = threadIdx.x;
  const float E = scal[2];
  float Tb = 0.f;
#pragma unroll
  for (int i = 0; i < NCHUNK; ++i) Tb += Tpart[b * NCHUNK + i];
  const float* kv = embk + 2 * OO;
  const float* bv = embb + 2 * OO;
  float p = 0.f;
  for (int o = tid; o < OO; o += 256) {
    float s = fmaf(kv[o], Tb, bv[o] * E);
    sv[o] = s;
    p += s;
  }
  red[tid] = p;
  __syncthreads();
  for (int s = 128; s > 0; s >>= 1) { if (tid < s) red[tid] += red[tid + s]; __syncthreads(); }
  if (tid == 0) s_u = red[0] / (float)OO;
  __syncthreads();
  const float u = s_u;
  float q = 0.f;
  for (int o = tid; o < OO; o += 256) { float d = sv[o] - u; q = fmaf(d, d, q); }
  red[tid] = q;
  __syncthreads();
  for (int s = 128; s > 0; s >>= 1) { if (tid < s) red[tid] += red[tid + s]; __syncthreads(); }
  if (tid == 0) s_inv = rsqrtf(red[0] / (float)OO + EPS_LN);
  __syncthreads();
  const float inv = s_inv;
  for (int o = tid; o < OO; o += 256)
    feat[(size_t)b * THREE_O + 2 * OO + o] = fmaf((sv[o] - u) * inv, ng[o], nb[o]);
}

// ---------------------------------------------------------------------------
// Final WMMA GEMM: out = feat[128,1536] @ k3[1536,512] + b3
// ---------------------------------------------------------------------------
__global__ __launch_bounds__(32) void gemm_out_kernel(
    const float* __restrict__ feat, const float* __restrict__ k3,
    const float* __restrict__ b3, float* __restrict__ out) {
  const int lane = threadIdx.x;
  const int r = lane & 15;
  const int h = lane >> 4;
  const int row = blockIdx.y * 16 + r;
  const int col = blockIdx.x * 16 + r;
  v8f acc = {};
  for (int k0 = 0; k0 < THREE_O; k0 += 4) {
    const int ka = k0 + 2 * h;
    v2f av = *(const v2f*)(feat + (size_t)row * THREE_O + ka);
    v2f bv;
    bv.x = k3[(size_t)(ka + 0) * OO + col];
    bv.y = k3[(size_t)(ka + 1) * OO + col];
    acc = __builtin_amdgcn_wmma_f32_16x16x4_f32(false, av, false, bv, (short)0, acc, false, false);
  }
  for (int i = 0; i < 8; ++i) {
    const int m = blockIdx.y * 16 + i + 8 * h;
    const int n = blockIdx.x * 16 + r;
    out[(size_t)m * OO + n] = acc[i] + b3[n];
  }
}

// ---------------------------------------------------------------------------
extern "C" void kernel_launch(void* const* d_in, const int* in_sizes, int n_in,
                              void* d_out, int out_size, void* d_ws, size_t ws_size,
                              hipStream_t stream) {
  const float* x    = (const float*)d_in[0];
  const float* gt   = (const float*)d_in[1];
  const float* wt   = (const float*)d_in[2];
  const float* mt   = (const float*)d_in[3];
  const float* k1   = (const float*)d_in[4];
  const float* b1   = (const float*)d_in[5];
  const float* k3   = (const float*)d_in[6];
  const float* b3   = (const float*)d_in[7];
  const float* embk = (const float*)d_in[8];
  const float* embb = (const float*)d_in[9];
  const float* ek   = (const float*)d_in[10];
  const float* ng   = (const float*)d_in[11];
  const float* nb   = (const float*)d_in[12];
  float* out = (float*)d_out;

  float* ws    = (float*)d_ws;
  float* wa    = ws;                          // D*O
  float* ax    = wa + (size_t)DD * OO;        // B*D
  float* feat  = ax + (size_t)BB * DD;        // B*3O
  float* Tpart = feat + (size_t)BB * THREE_O; // B*NCHUNK
  float* scal  = Tpart + (size_t)BB * NCHUNK; // 3

  prep_wa_kernel<<<(DD * OO + 255) / 256, 256, 0, stream>>>(wt, mt, wa, DD * OO);
  prep_ax_kernel<<<(BB * DD + 255) / 256, 256, 0, stream>>>(x, ax, BB * DD);
  scalars_kernel<<<1, 256, 0, stream>>>(embk, embb, ek, scal);
  gemm_fused_kernel<<<dim3(OO / 16, BB / 16), 32, 0, stream>>>(x, ax, k1, gt, wa, b1, feat);
  attn_T_kernel<<<dim3(BB, NCHUNK), 256, 0, stream>>>(x, ek, scal, Tpart);
  c4_kernel<<<BB, 256, 0, stream>>>(embk, embb, scal, Tpart, ng, nb, feat);
  gemm_out_kernel<<<dim3(OO / 16, BB / 16), 32, 0, stream>>>(feat, k3, b3, out);
}